// DistortedConv2D_36945308680543
// MI455X (gfx1250) — compile-verified
//
#include <hip/hip_runtime.h>

// DistortedConv2D for MI455X (gfx1250, wave32).
//
// out[b,o,ho,wo] = sum_{c,p,q} bilinear(x[b,c], grid[ho*3+p, wo*3+q]) * W[o,c,p,q] + bias[o]
//  => per output pixel: 32x144 GEMM row. We tile 16 pixels per wave and run the
//     144-deep reduction through V_WMMA_F32_16X16X4_F32 (exact fp32, 36 k-steps,
//     2 M-tiles of 16 output channels).
//
// Memory plan (MI455X: 23.3 TB/s HBM, 192MB L2):
//   - x (67.1MB) is repacked NCHW->NHWC into d_ws (67.1MB needed) so one bilinear
//     corner fetch = 64B contiguous (4x b128) instead of 16 strided scalar loads.
//   - grid (18.9MB) read once; sampling coords/weights for a tile's 144 grid points
//     are computed ONCE and reused across all 4 batches (grid is batch-shared).
//   - The 604MB sampled tensor is never materialized: blended taps go to a
//     per-wave LDS tile S[144][16] (per-wave LDS ops are in-order -> no barrier).
//   - HBM floor ~ (67.1 + 18.9 + 134)MB / 23.3TB/s ~= 9.4us; compute (9.7 GFLOP) is free.

#define BATCH 4
#define CIN   16
#define HH    512
#define WW    512
#define COUT  32
#define KDIM  144      // CIN * 3 * 3, also = 16 pixels * 9 taps per tile

typedef float v2f __attribute__((ext_vector_type(2)));
typedef float v4f __attribute__((ext_vector_type(4)));
typedef float v8f __attribute__((ext_vector_type(8)));

// ---------------------------------------------------------------------------
// Kernel 1: NCHW -> NHWC repack so gathers fetch 16 contiguous channels.
// ---------------------------------------------------------------------------
__global__ void __launch_bounds__(256)
nchw_to_nhwc_kernel(const float* __restrict__ x, float* __restrict__ xt) {
  int idx = blockIdx.x * blockDim.x + threadIdx.x;      // over B*H*W
  if (idx >= BATCH * HH * WW) return;
  int b   = idx >> 18;                                   // / (512*512)
  int pix = idx & (HH * WW - 1);
  const float* src = x + ((size_t)b * CIN) * (HH * WW) + pix;
  float v[CIN];
#pragma unroll
  for (int c = 0; c < CIN; ++c) v[c] = src[(size_t)c * (HH * WW)];
  v4f* dst = (v4f*)(xt + (size_t)idx * CIN);
#pragma unroll
  for (int g = 0; g < 4; ++g) {
    v4f t;
    t.x = v[g * 4 + 0]; t.y = v[g * 4 + 1];
    t.z = v[g * 4 + 2]; t.w = v[g * 4 + 3];
    dst[g] = t;
  }
}

// ---------------------------------------------------------------------------
// Kernel 2: fused grid-sample + block-conv as WMMA GEMM.
// 128 threads = 4 waves; each wave owns one 16-pixel tile (one ho row segment)
// and loops over the 4 batches reusing its cached sampling coords.
// ---------------------------------------------------------------------------
__global__ void __launch_bounds__(128)
distorted_conv_wmma_kernel(const float* __restrict__ xt,
                           const float* __restrict__ grid,
                           const float* __restrict__ wts,
                           const float* __restrict__ bias,
                           float* __restrict__ out) {
  __shared__ float Wlds[COUT * KDIM];       // 18,432 B, shared by all 4 waves
  __shared__ float Slds[4 * KDIM * 16];     // 36,864 B, one S[144][16] per wave

  // Stage weights: wts is [32][16][3][3] = [32][144] with k = c*9 + p*3 + q,
  // exactly the K ordering our GEMM uses.
  for (int i = threadIdx.x; i < COUT * KDIM; i += 128) Wlds[i] = wts[i];
  __syncthreads();

  const int lane = threadIdx.x & 31;
  const int wv   = threadIdx.x >> 5;
  float* S = &Slds[wv * (KDIM * 16)];

  const int tile = blockIdx.x * 4 + wv;     // 0..16383 pixel tiles
  const int ho   = tile >> 5;               // 32 tiles of 16 pixels per row
  const int wo0  = (tile & 31) << 4;

  // ---- Precompute the tile's 144 grid points (batch-invariant) -------------
  // g = i*32 + lane;  n = pixel-in-tile = g/9, tap = p*3+q = g%9.
  int   o00[5], o01[5], o10[5], o11[5], sb[5];
  float w00[5], w01[5], w10[5], w11[5];
#pragma unroll
  for (int i = 0; i < 5; ++i) {
    int g = i * 32 + lane;
    if (g < 144) {
      int n = g / 9, tap = g % 9;
      int p = tap / 3, q = tap % 3;
      int hk = ho * 3 + p;
      int wk = (wo0 + n) * 3 + q;
      float2 gp = ((const float2*)grid)[hk * (WW * 3) + wk];
      // align_corners=False unnormalize: x = (gx+1)*256 - 0.5
      float fx = gp.x * 256.0f + 255.5f;
      float fy = gp.y * 256.0f + 255.5f;
      float x0f = floorf(fx), y0f = floorf(fy);
      int ix0 = (int)x0f, iy0 = (int)y0f;
      float tx = fx - x0f, ty = fy - y0f;
      float ux = 1.0f - tx, uy = 1.0f - ty;
      bool vx0 = (ix0 >= 0) && (ix0 < WW);
      bool vx1 = (ix0 >= -1) && (ix0 < WW - 1);
      bool vy0 = (iy0 >= 0) && (iy0 < HH);
      bool vy1 = (iy0 >= -1) && (iy0 < HH - 1);
      w00[i] = uy * ux * (float)(vy0 && vx0);   // zero-padding folded into weight
      w01[i] = uy * tx * (float)(vy0 && vx1);
      w10[i] = ty * ux * (float)(vy1 && vx0);
      w11[i] = ty * tx * (float)(vy1 && vx1);
      int cx0 = min(max(ix0, 0), WW - 1), cx1 = min(max(ix0 + 1, 0), WW - 1);
      int cy0 = min(max(iy0, 0), HH - 1), cy1 = min(max(iy0 + 1, 0), HH - 1);
      o00[i] = (cy0 * WW + cx0) * CIN;  o01[i] = (cy0 * WW + cx1) * CIN;
      o10[i] = (cy1 * WW + cx0) * CIN;  o11[i] = (cy1 * WW + cx1) * CIN;
      sb[i] = tap * 16 + n;                    // S index base; full = c*144 + sb
    } else {
      sb[i] = -1;
      o00[i] = o01[i] = o10[i] = o11[i] = 0;
      w00[i] = w01[i] = w10[i] = w11[i] = 0.0f;
    }
  }

  const int mrow  = lane & 15;          // M row within 16x16 tile / B column N
  const int khalf = (lane >> 4) << 1;   // K sub-offset: 0 (lanes 0-15), 2 (16-31)

#pragma unroll 1
  for (int b = 0; b < BATCH; ++b) {
    const float* xb = xt + (size_t)b * (HH * WW * CIN);

    // ---- Gather + bilinear blend into S[144][16] (this wave only) ----------
#pragma unroll
    for (int i = 0; i < 5; ++i) {
      if (sb[i] >= 0) {
        const v4f* p00 = (const v4f*)(xb + o00[i]);
        const v4f* p01 = (const v4f*)(xb + o01[i]);
        const v4f* p10 = (const v4f*)(xb + o10[i]);
        const v4f* p11 = (const v4f*)(xb + o11[i]);
        float a = w00[i], bb = w01[i], c = w10[i], d = w11[i];
#pragma unroll
        for (int gch = 0; gch < 4; ++gch) {    // 4 channel groups of 4
          v4f v = p00[gch] * a + p01[gch] * bb + p10[gch] * c + p11[gch] * d;
          int base = (gch * 4) * KDIM + sb[i];
          S[base + 0 * KDIM] = v.x;
          S[base + 1 * KDIM] = v.y;
          S[base + 2 * KDIM] = v.z;
          S[base + 3 * KDIM] = v.w;
        }
      }
    }
    // Per-wave LDS ops complete in order; compiler fence only.
    asm volatile("" ::: "memory");

    // ---- 32x144 x 144x16 GEMM via V_WMMA_F32_16X16X4_F32 -------------------
    // D layout: VGPR r holds M = r + 8*(lane>>4), N = lane&15.
    v8f c0, c1;
#pragma unroll
    for (int r = 0; r < 8; ++r) {
      c0[r] = bias[r + 8 * (lane >> 4)];
      c1[r] = bias[16 + r + 8 * (lane >> 4)];
    }
#pragma unroll
    for (int ks = 0; ks < KDIM / 4; ++ks) {    // 36 k-steps of K=4
      int ka = ks * 4 + khalf;
      // A 16x4 layout: lane m=lane&15 holds {W[m][ka], W[m][ka+1]} (b64, 8B aligned)
      v2f a0 = *(const v2f*)&Wlds[mrow * KDIM + ka];
      v2f a1 = *(const v2f*)&Wlds[(mrow + 16) * KDIM + ka];
      // B 4x16 layout: lane n=lane&15 holds {S[ka][n], S[ka+1][n]}
      v2f bf;
      bf.x = S[ka * 16 + mrow];
      bf.y = S[(ka + 1) * 16 + mrow];
      c0 = __builtin_amdgcn_wmma_f32_16x16x4_f32(false, a0, false, bf,
                                                 (short)0, c0, false, false);
      c1 = __builtin_amdgcn_wmma_f32_16x16x4_f32(false, a1, false, bf,
                                                 (short)0, c1, false, false);
    }
    asm volatile("" ::: "memory");   // S will be overwritten by next batch

    // ---- Store: lanes 0-15 cover one 64B run per VGPR, lanes 16-31 another --
    float* ob = out + (((size_t)b * COUT) * HH + ho) * WW + wo0 + mrow;
#pragma unroll
    for (int r = 0; r < 8; ++r) {
      int o = r + 8 * (lane >> 4);
      ob[(size_t)o * (HH * WW)]        = c0[r];
      ob[(size_t)(o + 16) * (HH * WW)] = c1[r];
    }
  }
}

// ---------------------------------------------------------------------------
extern "C" void kernel_launch(void* const* d_in, const int* in_sizes, int n_in,
                              void* d_out, int out_size, void* d_ws, size_t ws_size,
                              hipStream_t stream) {
  const float* x    = (const float*)d_in[0];   // [4,16,512,512]
  const float* grid = (const float*)d_in[1];   // [1536,1536,2]
  const float* wts  = (const float*)d_in[2];   // [32,16,3,3]
  const float* bias = (const float*)d_in[3];   // [32]
  float*       out  = (float*)d_out;           // [4,32,512,512]
  float*       xt   = (float*)d_ws;            // NHWC repack, 67.1 MB

  (void)in_sizes; (void)n_in; (void)out_size; (void)ws_size;

  nchw_to_nhwc_kernel<<<(BATCH * HH * WW + 255) / 256, 256, 0, stream>>>(x, xt);

  // 16384 pixel tiles, 4 waves (tiles) per 128-thread block.
  distorted_conv_wmma_kernel<<<16384 / 4, 128, 0, stream>>>(xt, grid, wts, bias, out);
}